// StateBasedTokenizer_63814624084682
// MI455X (gfx1250) — compile-verified
//
#include <hip/hip_runtime.h>

// Problem constants (match reference)
#define Bc 32
#define Hc 1024
#define Nc 512
#define Ec 256
#define Vc 8192
#define Mc (Bc * Nc)   // 16384 rows

typedef float v2f __attribute__((ext_vector_type(2)));
typedef float v8f __attribute__((ext_vector_type(8)));
typedef int   v4i_vs __attribute__((vector_size(16)));   // matches builtin param

// ---------------------------------------------------------------------------
// CDNA5 async global->LDS copy (ASYNCcnt-tracked). Falls back to a synchronous
// float4 copy if the builtin is absent, so compilation never hard-fails.
// Builtin signature (from hipcc diagnostic): param0 = global int4*, then LDS
// dst, imm offset, imm cpol.
// ---------------------------------------------------------------------------
__device__ __forceinline__ void async_b128(void* lds_dst, const void* g_src) {
#if __has_builtin(__builtin_amdgcn_global_load_async_to_lds_b128)
    __builtin_amdgcn_global_load_async_to_lds_b128(
        (__attribute__((address_space(1))) v4i_vs*)(unsigned long long)g_src,
        (__attribute__((address_space(3))) v4i_vs*)(unsigned)(unsigned long long)lds_dst,
        0, 0);
#else
    *(float4*)lds_dst = *(const float4*)g_src;
#endif
}

__device__ __forceinline__ void wait_async0() {
#if __has_builtin(__builtin_amdgcn_s_wait_asynccnt)
    __builtin_amdgcn_s_wait_asynccnt(0);
#else
    asm volatile("s_wait_asynccnt 0x0" ::: "memory");
#endif
}

// ---------------------------------------------------------------------------
// Kernel 1: z[m,e] = sum_h z_enc[b,h,n] * w_pre[e,h] + b_pre[e]
// M-tile 128 (8 waves x 16), E-tile 64 (4 accumulators of 16 per wave).
// f32 WMMA 16x16x4; async double-buffered LDS staging.
// ---------------------------------------------------------------------------
#define AS_STRIDE 136   // mult of 4 (b128-aligned); 2*136%64=16 -> halves disjoint
#define BS_STRIDE 20    // mult of 4; gcd(20,64)=4, koff +/-2 -> conflict-free

__global__ __launch_bounds__(256) void k_pre_gemm(
    const float* __restrict__ z_enc, const float* __restrict__ w_pre,
    const float* __restrict__ b_pre, float* __restrict__ z_out)
{
    __shared__ float As[2][16 * AS_STRIDE];   // [k][m], m = 0..127
    __shared__ float Bs[2][64 * BS_STRIDE];   // [e][k], e = 0..63, k = 0..15

    const int t    = threadIdx.x;
    const int lane = t & 31;
    const int wave = t >> 5;
    const int half = lane >> 4;
    const int l16  = lane & 15;
    const int koff = half * 2;         // A/B VGPR K offset per ISA layout

    const int m0 = blockIdx.x * 128;   // m = b*N + n; 128 | 512 keeps one batch
    const int e0 = blockIdx.y * 64;
    const int batch = m0 / Nc;
    const int n0 = m0 % Nc;
    const float* zb = z_enc + (size_t)batch * Hc * Nc;

    // A stage: 16 rows x 128 floats = 512 x 16B chunks (2 per thread)
    auto stageA = [&](float* dstA, int k0) {
        #pragma unroll
        for (int i = 0; i < 2; ++i) {
            int c   = i * 256 + t;
            int row = c >> 5;            // 32 chunks per row
            int col = (c & 31) * 4;
            async_b128(dstA + row * AS_STRIDE + col,
                       zb + (size_t)(k0 + row) * Nc + n0 + col);
        }
    };
    // B stage: 64 rows x 16 floats = 256 x 16B chunks (1 per thread)
    auto stageB = [&](float* dstB, int k0) {
        int e  = t >> 2;
        int kp = (t & 3) * 4;
        async_b128(dstB + e * BS_STRIDE + kp,
                   w_pre + (size_t)(e0 + e) * Hc + k0 + kp);
    };

    v8f acc0 = {}, acc1 = {}, acc2 = {}, acc3 = {};

    stageA(As[0], 0);
    stageB(Bs[0], 0);
    wait_async0();
    __syncthreads();

    for (int k0 = 0; k0 < Hc; k0 += 16) {
        const int cur = (k0 >> 4) & 1;
        if (k0 + 16 < Hc) {            // prefetch next stage into other buffer
            stageA(As[cur ^ 1], k0 + 16);
            stageB(Bs[cur ^ 1], k0 + 16);
        }
        const float* Ab = As[cur];
        const float* Bb = Bs[cur];
        const int mb = wave * 16;
        #pragma unroll
        for (int k = 0; k < 16; k += 4) {
            v2f a;
            a.x = Ab[(k + koff    ) * AS_STRIDE + mb + l16];
            a.y = Ab[(k + koff + 1) * AS_STRIDE + mb + l16];
            v2f b0, b1, b2, b3;
            b0.x = Bb[(l16 +  0) * BS_STRIDE + k + koff];
            b0.y = Bb[(l16 +  0) * BS_STRIDE + k + koff + 1];
            b1.x = Bb[(l16 + 16) * BS_STRIDE + k + koff];
            b1.y = Bb[(l16 + 16) * BS_STRIDE + k + koff + 1];
            b2.x = Bb[(l16 + 32) * BS_STRIDE + k + koff];
            b2.y = Bb[(l16 + 32) * BS_STRIDE + k + koff + 1];
            b3.x = Bb[(l16 + 48) * BS_STRIDE + k + koff];
            b3.y = Bb[(l16 + 48) * BS_STRIDE + k + koff + 1];
            acc0 = __builtin_amdgcn_wmma_f32_16x16x4_f32(false, a, false, b0, (short)0, acc0, false, false);
            acc1 = __builtin_amdgcn_wmma_f32_16x16x4_f32(false, a, false, b1, (short)0, acc1, false, false);
            acc2 = __builtin_amdgcn_wmma_f32_16x16x4_f32(false, a, false, b2, (short)0, acc2, false, false);
            acc3 = __builtin_amdgcn_wmma_f32_16x16x4_f32(false, a, false, b3, (short)0, acc3, false, false);
        }
        wait_async0();                 // prefetched buffer complete
        __syncthreads();               // all waves done with buf[cur] + see next
    }

    // Epilogue: bias + store. C layout: VGPR r -> M = r (+8 for lanes 16-31).
    v8f accs[4] = {acc0, acc1, acc2, acc3};
    #pragma unroll
    for (int j = 0; j < 4; ++j) {
        int ecol = e0 + j * 16 + l16;
        float bias = b_pre[ecol];
        #pragma unroll
        for (int r = 0; r < 8; ++r) {
            int m = m0 + wave * 16 + r + half * 8;
            z_out[(size_t)m * Ec + ecol] = accs[j][r] + bias;
        }
    }
}

// ---------------------------------------------------------------------------
// Kernel 2: cnorm[v] = ||codebook[v]||^2   (one wave per row)
// ---------------------------------------------------------------------------
__global__ __launch_bounds__(256) void k_cnorm(
    const float* __restrict__ codebook, float* __restrict__ cnorm)
{
    const int wave = threadIdx.x >> 5;
    const int lane = threadIdx.x & 31;
    const int v = blockIdx.x * 8 + wave;
    const float* row = codebook + (size_t)v * Ec;
    float s = 0.f;
    #pragma unroll
    for (int i = 0; i < Ec / 32; ++i) {
        float x = row[lane + i * 32];
        s += x * x;
    }
    #pragma unroll
    for (int off = 16; off > 0; off >>= 1) s += __shfl_xor(s, off, 32);
    if (lane == 0) cnorm[v] = s;
}

// ---------------------------------------------------------------------------
// Kernel 3: fused dist GEMM + argmin.
// WG = 8 waves; wave w owns 16 rows (z tile resident in LDS).
// Per step: shared 32-row codebook tile (async double-buffered); each wave
// computes a 16x32 score tile (A fragment reused across 2 column tiles ->
// 1.5 ds-ops per WMMA), with running per-lane (min, argmin).
// ---------------------------------------------------------------------------
#define AZ_STRIDE 260   // mult of 4; lane bank = (4*lane + k) % 64, conflict-free
#define VT 32           // codebook rows per step

__global__ __launch_bounds__(256) void k_argmin(
    const float* __restrict__ z, const float* __restrict__ codebook,
    const float* __restrict__ cnorm, int* __restrict__ tok_ws,
    float* __restrict__ tok_out)
{
    extern __shared__ float smem[];
    float* Az  = smem;                           // [8][16][AZ_STRIDE]
    float* Cb0 = Az + 8 * 16 * AZ_STRIDE;        // [VT][AZ_STRIDE]
    float* Cb1 = Cb0 + VT * AZ_STRIDE;           // [VT][AZ_STRIDE]

    const int t    = threadIdx.x;
    const int lane = t & 31;
    const int wave = t >> 5;
    const int half = lane >> 4;
    const int l16  = lane & 15;
    const int koff = half * 2;

    const int m0 = (blockIdx.x * 8 + wave) * 16;
    float* Aw = Az + wave * 16 * AZ_STRIDE;

    // Stage this wave's 16 z rows (only this wave reads them back)
    for (int idx = lane; idx < 16 * Ec; idx += 32) {
        int m = idx >> 8;
        int e = idx & 255;
        Aw[m * AZ_STRIDE + e] = z[(size_t)(m0 + m) * Ec + e];
    }

    // Codebook stage: VT rows x 256 floats = 2048 x 16B chunks (8 per thread)
    auto stageC = [&](float* dst, int v0) {
        #pragma unroll
        for (int i = 0; i < 8; ++i) {
            int c   = i * 256 + t;
            int row = c >> 6;            // 64 chunks per row
            int col = (c & 63) * 4;
            async_b128(dst + row * AZ_STRIDE + col,
                       codebook + (size_t)(v0 + row) * Ec + col);
        }
    };

    float best[8];
    int   bidx[8];
    #pragma unroll
    for (int r = 0; r < 8; ++r) { best[r] = 3.4e38f; bidx[r] = 0; }

    stageC(Cb0, 0);
    wait_async0();
    __syncthreads();

    for (int v0 = 0; v0 < Vc; v0 += VT) {
        const int cur = (v0 / VT) & 1;
        float* Cb = cur ? Cb1 : Cb0;
        if (v0 + VT < Vc) stageC(cur ? Cb0 : Cb1, v0 + VT);  // async prefetch

        // S = Z(16x256) * C^T(256x32): A fragment feeds both column tiles
        v8f acc00 = {}, acc01 = {}, acc10 = {}, acc11 = {};
        #pragma unroll
        for (int k = 0; k < Ec; k += 8) {
            v2f a0, a1, b00, b01, b10, b11;
            a0.x  = Aw[l16 * AZ_STRIDE + k + koff];
            a0.y  = Aw[l16 * AZ_STRIDE + k + koff + 1];
            a1.x  = Aw[l16 * AZ_STRIDE + k + 4 + koff];
            a1.y  = Aw[l16 * AZ_STRIDE + k + 4 + koff + 1];
            b00.x = Cb[l16 * AZ_STRIDE + k + koff];
            b00.y = Cb[l16 * AZ_STRIDE + k + koff + 1];
            b01.x = Cb[l16 * AZ_STRIDE + k + 4 + koff];
            b01.y = Cb[l16 * AZ_STRIDE + k + 4 + koff + 1];
            b10.x = Cb[(l16 + 16) * AZ_STRIDE + k + koff];
            b10.y = Cb[(l16 + 16) * AZ_STRIDE + k + koff + 1];
            b11.x = Cb[(l16 + 16) * AZ_STRIDE + k + 4 + koff];
            b11.y = Cb[(l16 + 16) * AZ_STRIDE + k + 4 + koff + 1];
            acc00 = __builtin_amdgcn_wmma_f32_16x16x4_f32(false, a0, false, b00, (short)0, acc00, false, false);
            acc10 = __builtin_amdgcn_wmma_f32_16x16x4_f32(false, a0, false, b10, (short)0, acc10, false, false);
            acc01 = __builtin_amdgcn_wmma_f32_16x16x4_f32(false, a1, false, b01, (short)0, acc01, false, false);
            acc11 = __builtin_amdgcn_wmma_f32_16x16x4_f32(false, a1, false, b11, (short)0, acc11, false, false);
        }

        const int   vv0 = v0 + l16;
        const int   vv1 = v0 + 16 + l16;
        const float cn0 = cnorm[vv0];
        const float cn1 = cnorm[vv1];
        #pragma unroll
        for (int r = 0; r < 8; ++r) {
            float s0 = cn0 - 2.0f * (acc00[r] + acc01[r]);
            float s1 = cn1 - 2.0f * (acc10[r] + acc11[r]);
            if (s0 < best[r]) { best[r] = s0; bidx[r] = vv0; }
            if (s1 < best[r]) { best[r] = s1; bidx[r] = vv1; }
        }

        wait_async0();      // prefetched tile landed in LDS
        __syncthreads();    // all waves done with Cb + see next tile
    }

    // Reduce (min, argmin) across the 16 lanes of each half-wave
    #pragma unroll
    for (int off = 1; off < 16; off <<= 1) {
        #pragma unroll
        for (int r = 0; r < 8; ++r) {
            float ov = __shfl_xor(best[r], off, 32);
            int   oi = __shfl_xor(bidx[r], off, 32);
            if (ov < best[r] || (ov == best[r] && oi < bidx[r])) {
                best[r] = ov; bidx[r] = oi;
            }
        }
    }
    if (l16 == 0) {
        #pragma unroll
        for (int r = 0; r < 8; ++r) {
            int m = m0 + r + half * 8;
            tok_ws[m]  = bidx[r];
            tok_out[m] = (float)bidx[r];
        }
    }
}

// ---------------------------------------------------------------------------
// Kernel 4: z_q[m] = codebook[tokens[m]]  (wave per row, float4 copies)
// ---------------------------------------------------------------------------
__global__ __launch_bounds__(256) void k_gather(
    const float* __restrict__ codebook, const int* __restrict__ tok_ws,
    float* __restrict__ zq)
{
    const int wave = threadIdx.x >> 5;
    const int lane = threadIdx.x & 31;
    const int m = blockIdx.x * 8 + wave;
    const int tok = tok_ws[m];
    const float4* src = (const float4*)(codebook + (size_t)tok * Ec);
    float4* dst = (float4*)(zq + (size_t)m * Ec);
    dst[lane]      = src[lane];
    dst[lane + 32] = src[lane + 32];
}

// ---------------------------------------------------------------------------
extern "C" void kernel_launch(void* const* d_in, const int* in_sizes, int n_in,
                              void* d_out, int out_size, void* d_ws, size_t ws_size,
                              hipStream_t stream)
{
    (void)in_sizes; (void)n_in; (void)out_size; (void)ws_size;

    const float* z_enc    = (const float*)d_in[0];   // [B, H, N]
    const float* w_pre    = (const float*)d_in[1];   // [E, H]
    const float* b_pre    = (const float*)d_in[2];   // [E]
    const float* codebook = (const float*)d_in[3];   // [V, E]

    float* z_out   = (float*)d_out;                  // [M, E]
    float* zq_out  = z_out + (size_t)Mc * Ec;        // [M, E]
    float* tok_out = zq_out + (size_t)Mc * Ec;       // [M]

    float* cnorm  = (float*)d_ws;                                     // V floats
    int*   tok_ws = (int*)((char*)d_ws + (size_t)Vc * sizeof(float)); // M ints

    // 1) pre-quant conv as GEMM (fp32 WMMA, async double-buffered staging)
    k_pre_gemm<<<dim3(Mc / 128, Ec / 64), 256, 0, stream>>>(z_enc, w_pre, b_pre, z_out);

    // 2) codebook squared norms
    k_cnorm<<<dim3(Vc / 8), 256, 0, stream>>>(codebook, cnorm);

    // 3) fused distance GEMM + argmin (dynamic LDS: 8 z-tiles + 2 cb buffers)
    size_t lds3 = (size_t)(8 * 16 * AZ_STRIDE + 2 * VT * AZ_STRIDE) * sizeof(float);
    k_argmin<<<dim3(Mc / 128), 256, lds3, stream>>>(z_out, codebook, cnorm, tok_ws, tok_out);

    // 4) gather quantized rows
    k_gather<<<dim3(Mc / 8), 256, 0, stream>>>(codebook, tok_ws, zq_out);
}